// UpsampleBilinear3D_23905787969654
// MI455X (gfx1250) — compile-verified
//
#include <hip/hip_runtime.h>

typedef float v2f __attribute__((ext_vector_type(2)));
typedef float v8f __attribute__((ext_vector_type(8)));

namespace {
constexpr int kC   = 64;            // channels
constexpr int kDI  = 32;            // input extent per spatial dim
constexpr int kDO  = 64;            // output extent per spatial dim
constexpr int kWST = 72;            // LDS floats per w slot (64 + 8 pad -> conflict-free half-wave reads)
constexpr int kROW = 12 * kWST;     // floats per (d,h) row in LDS
constexpr int kLDS = 16 * kROW;     // 13824 floats = 54 KiB
}

// 1-D transposed-conv tap for scale-2 kernel [.25,.75,.75,.25].
// Output o (relative, 0..15) reads staged column j (0..11), where staged
// column j holds x[w_base - 1 + j].
__device__ __forceinline__ float tap_w(int o, int j) {
  const int a = o >> 1;
  float r = 0.0f;
  if ((o & 1) == 0) {
    if (j == a)     r = 0.25f;
    if (j == a + 1) r = 0.75f;
  } else {
    if (j == a + 1) r = 0.75f;
    if (j == a + 2) r = 0.25f;
  }
  return r;
}

__device__ __forceinline__ v8f fma8(v8f acc, float w, v8f y) {
#pragma unroll
  for (int i = 0; i < 8; ++i) acc[i] = __builtin_fmaf(w, y[i], acc[i]);
  return acc;
}

// Store one (d,h) output row: 16 w positions x 16 channels held in 8 VGPRs
// with the WMMA f32 C/D layout (VGPR r, lanes 0-15 -> w=r, lanes 16-31 -> w=r+8).
// Streamed output (no reuse): non-temporal so L2 stays reserved for the input.
__device__ __forceinline__ void store_row(float* __restrict__ out, size_t base,
                                          int hi, float wp, float wc,
                                          const v8f& P, const v8f& Cc) {
#pragma unroll
  for (int r = 0; r < 8; ++r) {
    __builtin_nontemporal_store(__builtin_fmaf(wp, P[r], wc * Cc[r]),
                                out + base + (size_t)((r + 8 * hi) * kC));
  }
}

__global__ void __launch_bounds__(128)
upsample_trilinear2x_wmma(const float* __restrict__ x, float* __restrict__ out) {
  __shared__ float xs[kLDS];

  // Block tile: 4(d) x 4(h) x 16(w) output voxels, all 64 channels.
  const int b  = blockIdx.x;
  const int wt = b & 3;          // 4 tiles of 16 along W
  const int ht = (b >> 2) & 15;  // 16 tiles of 4 along H
  const int dt = (b >> 6) & 15;  // 16 tiles of 4 along D
  const int n  = b >> 10;        // batch (2)

  const int ow0 = wt << 4;
  const int oh0 = ht << 2;
  const int od0 = dt << 2;
  const int wm0 = wt << 3;  // staged column j maps to input w = wm0 - 1 + j
  const int hm0 = ht << 1;  // staged row jh maps to input h = hm0 - 1 + jh
  const int dm0 = dt << 1;  // staged plane jd maps to input d = dm0 - 1 + jd

  // ---- Stage 4 x 4 x 12 x 64 halo tile into LDS with CDNA5 async direct
  // memory->LDS copies (no VGPR round-trip; tracked by ASYNCcnt). Addresses
  // are clamped at the volume edges; out-of-range columns are neutralized
  // later by zero A-weights / zero H,D taps, so clamped (finite) data is safe. ----
  for (int t = threadIdx.x; t < 16 * 12 * 16; t += 128) {
    const int c4  = t & 15;         // 16 float4 per voxel
    const int w   = (t >> 4) % 12;  // staged w column
    const int row = t / 192;        // (jd*4 + jh)
    int d_in = dm0 - 1 + (row >> 2);
    int h_in = hm0 - 1 + (row & 3);
    int w_in = wm0 - 1 + w;
    d_in = d_in < 0 ? 0 : (d_in > kDI - 1 ? kDI - 1 : d_in);
    h_in = h_in < 0 ? 0 : (h_in > kDI - 1 ? kDI - 1 : h_in);
    w_in = w_in < 0 ? 0 : (w_in > kDI - 1 ? kDI - 1 : w_in);
    const float* gp = x + ((((size_t)n * kDI + d_in) * kDI + h_in) * kDI + w_in) * kC + c4 * 4;
    // Generic pointers to LDS carry the LDS-relative offset in the low 32 bits.
    const unsigned lp = (unsigned)(size_t)(xs + row * kROW + w * kWST + c4 * 4);
    asm volatile("global_load_async_to_lds_b128 %0, %1, off"
                 :
                 : "v"(lp), "v"(gp)
                 : "memory");
  }
  asm volatile("s_wait_asynccnt 0" ::: "memory");
  __syncthreads();

  const int lane = threadIdx.x & 31;
  const int c0   = (threadIdx.x >> 5) << 4;  // 16-channel chunk per wave
  const int lm   = lane & 15;
  const int hi   = lane >> 4;
  const int cidx = c0 + lm;

  // ---- Constant A operands (16x12 band matrix as 3 K=4 chunks).
  // f32 A layout: VGPR0 -> K = kb + 2*hi, VGPR1 -> K = kb + 1 + 2*hi; M = lane&15.
  // Fold the W-dim zero-padding boundary into A. ----
  v2f A[3];
#pragma unroll
  for (int cnk = 0; cnk < 3; ++cnk) {
    const int k0 = 4 * cnk + 2 * hi;
    float a0 = tap_w(lm, k0);
    float a1 = tap_w(lm, k0 + 1);
    const int wi0 = wm0 - 1 + k0;
    const int wi1 = wi0 + 1;
    if (wi0 < 0 || wi0 >= kDI) a0 = 0.0f;
    if (wi1 < 0 || wi1 >= kDI) a1 = 0.0f;
    A[cnk].x = a0;
    A[cnk].y = a1;
  }

  const v8f zero8 = {0.f, 0.f, 0.f, 0.f, 0.f, 0.f, 0.f, 0.f};
  v8f Zp0 = zero8, Zp1 = zero8, Zp2 = zero8, Zp3 = zero8;  // previous D plane

#pragma unroll
  for (int jd = 0; jd < 4; ++jd) {
    const int   d_in = dm0 - 1 + jd;
    const float dv   = (d_in >= 0 && d_in < kDI) ? 1.0f : 0.0f;

    // Current D plane: 4 H-output rows, each 16w x 16c in 8 VGPRs.
    v8f Z0 = zero8, Z1 = zero8, Z2 = zero8, Z3 = zero8;

#pragma unroll
    for (int jh = 0; jh < 4; ++jh) {
      const int   h_in = hm0 - 1 + jh;
      const float rv   = (h_in >= 0 && h_in < kDI) ? dv : 0.0f;  // uniform row validity
      const float* rp  = xs + (((jd << 2) | jh) * kROW) + c0 + lm;

      // W-dim upsample as GEMM: Y(16w x 16c) = A(16x12) * X(12 x 16c).
      // EXEC is all-ones here by construction (no divergent guards).
      v8f y = zero8;
#pragma unroll
      for (int cnk = 0; cnk < 3; ++cnk) {
        const int kb = 4 * cnk + 2 * hi;
        v2f Bv;
        Bv.x = rp[kb * kWST];        // B VGPR0: K = kb + 2*hi, N = lane&15
        Bv.y = rp[(kb + 1) * kWST];  // B VGPR1: K = kb + 1 + 2*hi
        y = __builtin_amdgcn_wmma_f32_16x16x4_f32(false, A[cnk], false, Bv,
                                                  (short)0, y, false, false);
      }

      // H-dim 2-tap combine, validity folded into the (uniform) tap weights.
      if (jh == 0) { Z0 = fma8(Z0, 0.25f * rv, y); }
      if (jh == 1) { Z0 = fma8(Z0, 0.75f * rv, y);
                     Z1 = fma8(Z1, 0.75f * rv, y);
                     Z2 = fma8(Z2, 0.25f * rv, y); }
      if (jh == 2) { Z1 = fma8(Z1, 0.25f * rv, y);
                     Z2 = fma8(Z2, 0.75f * rv, y);
                     Z3 = fma8(Z3, 0.75f * rv, y); }
      if (jh == 3) { Z3 = fma8(Z3, 0.25f * rv, y); }
    }

    // ---- D-dim 2-tap combine across consecutive staged planes; emit outputs. ----
    auto plane_base = [&](int td, int th) -> size_t {
      return ((((size_t)n * kDO + (od0 + td)) * kDO + (oh0 + th)) * kDO + ow0) * kC + cidx;
    };
    if (jd == 1) {  // out plane td=0 : 0.25*Z(jd0) + 0.75*Z(jd1)
      store_row(out, plane_base(0, 0), hi, 0.25f, 0.75f, Zp0, Z0);
      store_row(out, plane_base(0, 1), hi, 0.25f, 0.75f, Zp1, Z1);
      store_row(out, plane_base(0, 2), hi, 0.25f, 0.75f, Zp2, Z2);
      store_row(out, plane_base(0, 3), hi, 0.25f, 0.75f, Zp3, Z3);
    }
    if (jd == 2) {  // td=1 : 0.75*Z1 + 0.25*Z2 ; td=2 : 0.25*Z1 + 0.75*Z2
      store_row(out, plane_base(1, 0), hi, 0.75f, 0.25f, Zp0, Z0);
      store_row(out, plane_base(1, 1), hi, 0.75f, 0.25f, Zp1, Z1);
      store_row(out, plane_base(1, 2), hi, 0.75f, 0.25f, Zp2, Z2);
      store_row(out, plane_base(1, 3), hi, 0.75f, 0.25f, Zp3, Z3);
      store_row(out, plane_base(2, 0), hi, 0.25f, 0.75f, Zp0, Z0);
      store_row(out, plane_base(2, 1), hi, 0.25f, 0.75f, Zp1, Z1);
      store_row(out, plane_base(2, 2), hi, 0.25f, 0.75f, Zp2, Z2);
      store_row(out, plane_base(2, 3), hi, 0.25f, 0.75f, Zp3, Z3);
    }
    if (jd == 3) {  // td=3 : 0.75*Z2 + 0.25*Z3
      store_row(out, plane_base(3, 0), hi, 0.75f, 0.25f, Zp0, Z0);
      store_row(out, plane_base(3, 1), hi, 0.75f, 0.25f, Zp1, Z1);
      store_row(out, plane_base(3, 2), hi, 0.75f, 0.25f, Zp2, Z2);
      store_row(out, plane_base(3, 3), hi, 0.75f, 0.25f, Zp3, Z3);
    }
    Zp0 = Z0; Zp1 = Z1; Zp2 = Z2; Zp3 = Z3;
  }
}

extern "C" void kernel_launch(void* const* d_in, const int* in_sizes, int n_in,
                              void* d_out, int out_size, void* d_ws, size_t ws_size,
                              hipStream_t stream) {
  (void)in_sizes; (void)n_in; (void)out_size; (void)d_ws; (void)ws_size;
  const float* x = (const float*)d_in[0];   // (2,32,32,32,64) f32
  // d_in[1] (upsample_filter) is deterministic trilinear taps; baked into the kernel.
  float* out = (float*)d_out;               // (2,64,64,64,64) f32

  // Grid: 2(batch) * 16(D/4) * 16(H/4) * 4(W/16) tiles, 4 waves (128 thr) each.
  upsample_trilinear2x_wmma<<<2048, 128, 0, stream>>>(x, out);
}